// Correlation_6906307412101
// MI455X (gfx1250) — compile-verified
//
#include <hip/hip_runtime.h>
#include <hip/hip_bf16.h>

typedef float v2f __attribute__((ext_vector_type(2)));
typedef float v8f __attribute__((ext_vector_type(8)));

#define NATOM 16
#define FEAT  128
#define WAVES_PER_BLOCK 8
// per-wave LDS slab layout (floats):
//   S   : 16x17 matrix (272)
//   Vv  : 16   (272..287)
//   Vw  : 16   (288..303)
//   Vd  : 16   (304..319)
//   Ve  : 16   (320..335)
#define SLAB 336

__device__ __forceinline__ float waveReduceSum(float v) {
    v += __shfl_xor(v, 16, 32);
    v += __shfl_xor(v, 8, 32);
    v += __shfl_xor(v, 4, 32);
    v += __shfl_xor(v, 2, 32);
    v += __shfl_xor(v, 1, 32);
    return v;
}

__global__ __launch_bounds__(256) void min_eig_kernel(const float* __restrict__ sr,
                                                      float* __restrict__ out,
                                                      int M) {
    __shared__ float smem[WAVES_PER_BLOCK * SLAB];
    const int lane = threadIdx.x & 31;
    const int wid  = threadIdx.x >> 5;
    const int m_raw = blockIdx.x * WAVES_PER_BLOCK + wid;
    const int m = (m_raw < M) ? m_raw : (M - 1);   // clamp: keep syncs uniform

    float* S  = smem + wid * SLAB;
    float* Vv = S + 272;
    float* Vw = S + 288;
    float* Vd = S + 304;
    float* Ve = S + 320;

    // ---------------- Phase 1: C = T * T^T via v_wmma_f32_16x16x4_f32 ----
    // A (16x4 f32): lanes 0-15 -> M, VGPR0/1 = K{0,1}; lanes 16-31 -> K{2,3}.
    // B (4x16) is the lane-transposed layout, so SRC1 = SRC0 yields Gram.
    v8f c = {0.f,0.f,0.f,0.f,0.f,0.f,0.f,0.f};
    {
        const int r = lane & 15;      // matrix row this lane supplies
        const int h = lane >> 4;      // K-half
        const float* p = sr + (size_t)m * (NATOM * FEAT) + r * FEAT + 2 * h;
        #pragma unroll
        for (int k0 = 0; k0 < FEAT; k0 += 4) {
            v2f a = *(const v2f*)(p + k0);   // 8B-aligned b64 load
            c = __builtin_amdgcn_wmma_f32_16x16x4_f32(
                    false, a, false, a, (short)0, c, false, false);
        }
    }
    // C/D layout: VGPR j: lanes 0-15 -> M=j, N=lane; lanes 16-31 -> M=j+8.
    {
        const int n = lane & 15;
        const int rb = (lane >> 4) * 8;
        #pragma unroll
        for (int j = 0; j < 8; ++j) S[(rb + j) * 17 + n] = c[j];
    }
    __syncthreads();

    // ---------------- Phase 2: Householder tridiagonalization -----------
    // Lanes 0..15 each own one row of S. 14 reflections.
    const int i = lane;
    const bool act = (lane < NATOM);
    for (int k = 0; k <= NATOM - 3; ++k) {
        float xi = 0.f;
        if (act && i > k) xi = S[i * 17 + k];
        const float x1   = __shfl(xi, k + 1, 32);
        const float nrm2 = waveReduceSum(xi * xi);
        const float nrm  = sqrtf(nrm2);
        const float alpha = (x1 >= 0.f) ? -nrm : nrm;

        float vi = 0.f;
        if (act && i > k) vi = (i == k + 1) ? (x1 - alpha) : xi;
        if (act) Vv[i] = vi;
        if (lane == 0) Ve[k] = alpha;
        const float vtv = waveReduceSum(vi * vi);
        __syncthreads();

        const float beta = (vtv > 1e-30f) ? 2.f / vtv : 0.f;
        float pi = 0.f;
        if (act && i > k) {
            float acc = 0.f;
            for (int j = k + 1; j < NATOM; ++j) acc += S[i * 17 + j] * Vv[j];
            pi = beta * acc;
        }
        const float vtp = waveReduceSum(vi * pi);
        const float Kc  = 0.5f * beta * vtp;
        const float wi  = pi - Kc * vi;
        if (act) Vw[i] = wi;
        __syncthreads();

        if (act && i > k) {
            for (int j = k + 1; j < NATOM; ++j)
                S[i * 17 + j] -= vi * Vw[j] + wi * Vv[j];
        }
        __syncthreads();
    }
    if (act) Vd[i] = S[i * 17 + i];
    if (lane == 0) Ve[NATOM - 2] = S[(NATOM - 1) * 17 + (NATOM - 2)];
    __syncthreads();

    // ---------------- Phase 3: 32-way bisection with Sturm counts --------
    float d[NATOM], e[NATOM - 1];
    #pragma unroll
    for (int t = 0; t < NATOM; ++t) d[t] = Vd[t];
    #pragma unroll
    for (int t = 0; t < NATOM - 1; ++t) e[t] = Ve[t];

    // Gershgorin bracket
    float lo, hi;
    {
        float r0 = fabsf(e[0]);
        lo = d[0] - r0;
        hi = d[0] + r0;
        #pragma unroll
        for (int t = 1; t < NATOM; ++t) {
            float r = fabsf(e[t - 1]) + ((t < NATOM - 1) ? fabsf(e[t]) : 0.f);
            lo = fminf(lo, d[t] - r);
            hi = fmaxf(hi, d[t] + r);
        }
    }

    // Each round: 32 probe points partition [lo,hi] into 33 slices -> 33x shrink.
    for (int iter = 0; iter < 7; ++iter) {
        const float stepw = (hi - lo) * (1.0f / 33.0f);
        const float x = lo + stepw * (float)(lane + 1);
        // Sturm / LDL inertia: number of eigenvalues < x
        int cnt = 0;
        float q = d[0] - x;
        if (q < 0.f) cnt++;
        #pragma unroll
        for (int t = 1; t < NATOM; ++t) {
            if (q == 0.f) q = -1e-30f;
            q = (d[t] - x) - (e[t - 1] * e[t - 1]) / q;
            if (q < 0.f) cnt++;
        }
        const unsigned long long bal = __ballot(cnt == 0);
        const int nz = __popcll(bal & 0xffffffffull); // lanes still left of lambda_min
        const float nlo = lo + stepw * (float)nz;
        const float nhi = (nz >= 32) ? hi : (lo + stepw * (float)(nz + 1));
        lo = nlo;
        hi = nhi;
    }

    if (lane == 0 && m_raw < M) out[m_raw] = 0.5f * (lo + hi);
}

extern "C" void kernel_launch(void* const* d_in, const int* in_sizes, int n_in,
                              void* d_out, int out_size, void* d_ws, size_t ws_size,
                              hipStream_t stream) {
    const float* sr = (const float*)d_in[0];   // [M*16, 128] fp32
    // d_in[1] = idx_m (sorted, uniform segments) -> M == out_size, n == 16
    float* out = (float*)d_out;                // [M] fp32 min eigenvalues
    const int M = out_size;
    const int blocks = (M + WAVES_PER_BLOCK - 1) / WAVES_PER_BLOCK;
    min_eig_kernel<<<blocks, 256, 0, stream>>>(sr, out, M);
}